// DiscriminatorV1_91087666414033
// MI455X (gfx1250) — compile-verified
//
#include <hip/hip_runtime.h>
#include <hip/hip_fp16.h>
#include <math.h>

#define T_STEPS 100000
#define NOTE 215
#define STATE 200
#define MTILES (T_STEPS / 16)          // 6250
#define WAVES_PER_BLK 8

typedef __attribute__((ext_vector_type(2))) float v2f;
typedef __attribute__((ext_vector_type(8))) float v8f;

// ---------------------------------------------------------------------------
// Stage 1: f32 WMMA GEMMs over the whole track.
//   z=0: pre_f = track @ Wf[:215] + bf
//   z=1: pre_i = track @ Wi[:215] + bi
//   z=2: cand  = tanh(track @ Wc + bc)
// One wave computes one 16(M)x16(N) tile. Main K loop (k<212) has NO guards;
// out-of-range N columns use clamped addresses (their D columns are never
// stored, and B column n only influences D column n). K tail peeled once.
// ---------------------------------------------------------------------------
__global__ void precompute_gemm(const float* __restrict__ track,
                                const float* __restrict__ Wf,
                                const float* __restrict__ bf,
                                const float* __restrict__ Wi,
                                const float* __restrict__ bi,
                                const float* __restrict__ Wc,
                                const float* __restrict__ bc,
                                float* __restrict__ pf,
                                float* __restrict__ pi,
                                float* __restrict__ cd)
{
    const int z = blockIdx.z;
    const float* W;
    const float* bias;
    float* out;
    if (z == 0)      { W = Wf; bias = bf; out = pf; }
    else if (z == 1) { W = Wi; bias = bi; out = pi; }
    else             { W = Wc; bias = bc; out = cd; }

    const int mt = blockIdx.x * WAVES_PER_BLK + threadIdx.y;  // M-tile index
    if (mt >= MTILES) return;

    const int m0   = mt * 16;
    const int n0   = blockIdx.y * 16;
    const int lane = threadIdx.x;      // wave32
    const int half = lane >> 4;        // 0 -> K{+0,+1}, 1 -> K{+2,+3}
    const int l    = lane & 15;

    const int realcol = n0 + l;
    const int col     = (realcol < STATE) ? realcol : (STATE - 1);  // clamped addr

    const float* arow = track + (size_t)(m0 + l) * NOTE;
    const float* bcol = W + col;

    v8f acc = {};
    // Main loop: k0 = 0,4,...,208 -> ka,ka+1 <= 211 < 215. No guards anywhere.
    for (int k0 = 0; k0 < 212; k0 += 4) {
        const int ka = k0 + half * 2;
        v2f a, b;
        a.x = arow[ka];
        a.y = arow[ka + 1];
        b.x = bcol[(size_t)ka * STATE];
        b.y = bcol[(size_t)(ka + 1) * STATE];
        acc = __builtin_amdgcn_wmma_f32_16x16x4_f32(false, a, false, b,
                                                    (short)0, acc, false, false);
    }
    // Tail: k = 212..214. ka = 212 or 214 (always valid); ka+1 = 213 or 215.
    {
        const int ka   = 212 + half * 2;
        const int kb   = ka + 1;
        const int kbs  = (kb < NOTE) ? kb : (NOTE - 1);   // safe address
        const bool kok = (kb < NOTE);
        v2f a, b;
        a.x = arow[ka];
        a.y = kok ? arow[kbs] : 0.0f;          // unconditional-safe addr + select
        b.x = bcol[(size_t)ka * STATE];
        {
            const float by = bcol[(size_t)kbs * STATE];
            b.y = kok ? by : 0.0f;
        }
        acc = __builtin_amdgcn_wmma_f32_16x16x4_f32(false, a, false, b,
                                                    (short)0, acc, false, false);
    }

    if (realcol < STATE) {
        const float bv    = bias[realcol];
        const int   rbase = m0 + half * 8;     // C/D layout: lanes16-31 -> M=8..15
        #pragma unroll
        for (int i = 0; i < 8; ++i) {
            float v = acc[i] + bv;
            if (z == 2) v = tanhf(v);
            out[(size_t)(rbase + i) * STATE + realcol] = v;
        }
    }
}

// ---------------------------------------------------------------------------
// Stage 2: sequential scan, single workgroup (256 thr = 8 waves).
// Recurrent weights in LDS, packed 2 k-values per 8-byte element:
//   sW[k2*STATE + col] = { half2(wf[2k2],wi[2k2]), half2(wf[2k2+1],wi[2k2+1]) }
// -> one ds_load_b64 per 2 k's, conflict-free (32 lanes x 8B = 64 banks).
// State read via aligned float4 (ds_load_b128). One barrier per step via
// ping-pong state buffers.
// ---------------------------------------------------------------------------
struct __align__(8) WPair { __half2 a, b; };

__global__ void __launch_bounds__(256, 1)
scan_kernel(const float* __restrict__ Wf,   // (415,200); rows 215.. = recurrent
            const float* __restrict__ Wi,
            const float* __restrict__ state0,
            const float* __restrict__ PF,
            const float* __restrict__ PI,
            const float* __restrict__ CD,
            const float* __restrict__ W1, const float* __restrict__ b1,
            const float* __restrict__ W2, const float* __restrict__ b2,
            float* __restrict__ outv)
{
    __shared__ WPair sW[(STATE / 2) * STATE];          // 160000 B
    __shared__ __align__(16) float sbuf[2][STATE];     // ping-pong state
    __shared__ float hbuf[STATE];                      // head scratch
    __shared__ float red[2];                           // log_softmax reduction

    const int tid = threadIdx.x;

    // Preload recurrent weights (rows 215..414), packed to f16 pairs.
    for (int idx = tid; idx < (STATE / 2) * STATE; idx += blockDim.x) {
        const int k2 = idx / STATE;
        const int c  = idx - k2 * STATE;
        const int r0 = (NOTE + 2 * k2) * STATE + c;
        const int r1 = r0 + STATE;
        WPair p;
        p.a = __halves2half2(__float2half(Wf[r0]), __float2half(Wi[r0]));
        p.b = __halves2half2(__float2half(Wf[r1]), __float2half(Wi[r1]));
        sW[idx] = p;
    }
    if (tid < STATE) sbuf[0][tid] = state0[tid];
    __syncthreads();

    const int  col = tid;
    const bool ok  = (col < STATE);

    for (int t = 0; t < T_STEPS; ++t) {
        const float* scur = sbuf[t & 1];

        if (ok) {
            const float pfv = PF[(size_t)t * STATE + col];
            const float piv = PI[(size_t)t * STATE + col];
            const float cv  = CD[(size_t)t * STATE + col];
            if (t + 32 < T_STEPS) {
                __builtin_prefetch(&PF[(size_t)(t + 32) * STATE + col], 0, 0);
                __builtin_prefetch(&PI[(size_t)(t + 32) * STATE + col], 0, 0);
                __builtin_prefetch(&CD[(size_t)(t + 32) * STATE + col], 0, 0);
            }

            // 4 accumulator chains per gate for ILP.
            float af0 = 0.f, af1 = 0.f, af2 = 0.f, af3 = 0.f;
            float ai0 = 0.f, ai1 = 0.f, ai2 = 0.f, ai3 = 0.f;
            const WPair* w = sW + col;
            for (int k2 = 0; k2 < STATE / 2; k2 += 2) {
                const float4 sv = *(const float4*)(scur + 2 * k2);
                const WPair p0 = w[(k2    ) * STATE];   // k = 2k2, 2k2+1
                const WPair p1 = w[(k2 + 1) * STATE];   // k = 2k2+2, 2k2+3
                const float2 a0 = __half22float2(p0.a);
                const float2 a1 = __half22float2(p0.b);
                const float2 a2 = __half22float2(p1.a);
                const float2 a3 = __half22float2(p1.b);
                af0 = fmaf(a0.x, sv.x, af0); ai0 = fmaf(a0.y, sv.x, ai0);
                af1 = fmaf(a1.x, sv.y, af1); ai1 = fmaf(a1.y, sv.y, ai1);
                af2 = fmaf(a2.x, sv.z, af2); ai2 = fmaf(a2.y, sv.z, ai2);
                af3 = fmaf(a3.x, sv.w, af3); ai3 = fmaf(a3.y, sv.w, ai3);
            }
            const float dotf = (af0 + af1) + (af2 + af3);
            const float doti = (ai0 + ai1) + (ai2 + ai3);
            const float f = 1.0f / (1.0f + __expf(-(pfv + dotf)));
            const float g = 1.0f / (1.0f + __expf(-(piv + doti)));
            sbuf[(t + 1) & 1][col] = scur[col] * f + g * cv;
        }
        __syncthreads();
    }

    // ---- output head: relu(L2(relu(L1(s)))) then log_softmax ----
    const float* sf = sbuf[T_STEPS & 1];
    if (ok) {
        float a = b1[col];
        for (int k = 0; k < STATE; ++k) a = fmaf(sf[k], W1[k * STATE + col], a);
        hbuf[col] = a > 0.0f ? a : 0.0f;
    }
    __syncthreads();

    float h2 = 0.0f;
    if (ok) {
        float a = b2[col];
        for (int k = 0; k < STATE; ++k) a = fmaf(hbuf[k], W2[k * STATE + col], a);
        h2 = a > 0.0f ? a : 0.0f;
        sbuf[1][col] = h2;
    }
    __syncthreads();

    if (tid == 0) {
        float m = -__builtin_inff();
        for (int k = 0; k < STATE; ++k) m = fmaxf(m, sbuf[1][k]);
        float s = 0.0f;
        for (int k = 0; k < STATE; ++k) s += __expf(sbuf[1][k] - m);
        red[0] = m;
        red[1] = logf(s);
    }
    __syncthreads();

    if (ok) outv[col] = h2 - red[0] - red[1];
}

// ---------------------------------------------------------------------------
extern "C" void kernel_launch(void* const* d_in, const int* in_sizes, int n_in,
                              void* d_out, int out_size, void* d_ws, size_t ws_size,
                              hipStream_t stream)
{
    const float* track  = (const float*)d_in[0];
    const float* state0 = (const float*)d_in[1];
    const float* Wf     = (const float*)d_in[2];
    const float* bf     = (const float*)d_in[3];
    const float* Wi     = (const float*)d_in[4];
    const float* bi     = (const float*)d_in[5];
    const float* Wc     = (const float*)d_in[6];
    const float* bc     = (const float*)d_in[7];
    const float* W1     = (const float*)d_in[8];
    const float* b1     = (const float*)d_in[9];
    const float* W2     = (const float*)d_in[10];
    const float* b2     = (const float*)d_in[11];
    float* out = (float*)d_out;

    // Workspace: pre_f | pre_i | cand  (3 * 100000 * 200 f32 = 240 MB)
    float* pf = (float*)d_ws;
    float* pi = pf + (size_t)T_STEPS * STATE;
    float* cd = pi + (size_t)T_STEPS * STATE;

    dim3 grid((MTILES + WAVES_PER_BLK - 1) / WAVES_PER_BLK,   // 782
              (STATE + 15) / 16,                              // 13
              3);
    dim3 blk(32, WAVES_PER_BLK);
    precompute_gemm<<<grid, blk, 0, stream>>>(track, Wf, bf, Wi, bi, Wc, bc,
                                              pf, pi, cd);
    scan_kernel<<<1, 256, 0, stream>>>(Wf, Wi, state0, pf, pi, cd,
                                       W1, b1, W2, b2, out);
}